// JonesModel_14181982011922
// MI455X (gfx1250) — compile-verified
//
#include <hip/hip_runtime.h>
#include <stdint.h>

// ---------------------------------------------------------------------------
// JonesModel: V_p[a,b,v,f] = sum_{c,d} J[a,c,ant1[v],f] * V[c,d,v,f] * conj(J[b,d,ant2[v],f])
// Memory-bound (~2 FLOP/byte): optimize the CDNA5 data path, not matrix ops.
//   - TDM (tensor_load_to_lds) stages the 4 complex planes of V[:,:,v,:,:]
//     (32 KB) into LDS with one async DMA per block; jones loads for the
//     first half of the freq range are issued *before* the tensor wait so
//     their latency overlaps the DMA.
//   - compute reads V via ds_load_b128 (LDS address space).
//   - jones (4.3 MB) stays L2-resident; plain global_load_b128.
//   - output streamed with non-temporal b128 stores.
// ---------------------------------------------------------------------------

typedef __attribute__((ext_vector_type(2))) float fvec2;
typedef __attribute__((ext_vector_type(4))) float fvec4;
typedef __attribute__((ext_vector_type(4))) unsigned int v4u;
typedef __attribute__((ext_vector_type(4))) int v4i;
typedef __attribute__((ext_vector_type(8))) int v8i;

#define NPOL  2
#define NANT  128
#define NVIS  8128
#define NFREQ 1024
#define V_PLANE  (NVIS * NFREQ * 2)   // 16,646,144 floats per (c,d) plane of V / out
#define J_PLANE  (NANT * NFREQ * 2)   // 262,144 floats per (p,q) plane of jones
#define ROW_F    (NFREQ * 2)          // 2048 floats per (vis or ant) row

static __device__ __forceinline__ fvec2 cmul(fvec2 a, fvec2 b) {   // a * b
    fvec2 r;
    r.x = fmaf(a.x, b.x, -(a.y * b.y));
    r.y = fmaf(a.x, b.y,  (a.y * b.x));
    return r;
}
static __device__ __forceinline__ fvec2 cmulc(fvec2 a, fvec2 b) {  // a * conj(b)
    fvec2 r;
    r.x = fmaf(a.x, b.x,  (a.y * b.y));
    r.y = fmaf(a.y, b.x, -(a.x * b.y));
    return r;
}
static __device__ __forceinline__ fvec2 cadd(fvec2 a, fvec2 b) {
    fvec2 r; r.x = a.x + b.x; r.y = a.y + b.y; return r;
}

__global__ __launch_bounds__(256) void jones_apply_kernel(
    const float* __restrict__ V,      // [2][2][NVIS][NFREQ][2]
    const float* __restrict__ jones,  // [2][2][NANT][NFREQ][2]
    const int*   __restrict__ ant1,   // [NVIS]
    const int*   __restrict__ ant2,   // [NVIS]
    float*       __restrict__ out)    // [2][2][NVIS][NFREQ][2]
{
    __shared__ float smem[4 * ROW_F];           // 32 KB: [cd][f*2 + ri]
    const int v = blockIdx.x;

    // ---- TDM: kick off the async DMA first so it runs under the jones loads.
    // 2D tile: 4 rows (c,d planes) x 2048 floats, row stride = plane stride.
    if (threadIdx.x < 32) {                     // wave 0 only (wave-uniform branch)
        const uint64_t gaddr = (uint64_t)(uintptr_t)(V + (size_t)v * ROW_F);
        const uint32_t laddr = (uint32_t)(uintptr_t)(void*)smem;  // LDS byte offset

        v4u g0;                                  // D# group 0 (128b)
        g0.x = 1u;                               // count=1 (valid), user mode
        g0.y = laddr;                            // lds_addr [63:32]
        g0.z = (uint32_t)(gaddr & 0xFFFFFFFFu);  // global_addr [95:64]
        g0.w = (uint32_t)((gaddr >> 32) & 0x01FFFFFFu) | (2u << 30); // addr hi | type=2

        v8i g1 = {                               // D# group 1 (256b)
            (int)0x00020000,                     // wg_mask=0, data_size=2 (4B)
            (int)0x08000000,                     // tensor_dim0 = 2048 (bits 79:48 lo16)
            (int)0x00040000,                     // tensor_dim0 hi=0 | tensor_dim1=4 lo16
            (int)0x08000000,                     // tensor_dim1 hi=0 | tile_dim0 = 2048
            (int)0x00000004,                     // tile_dim1 = 4, tile_dim2 = 0
            (int)V_PLANE,                        // tensor_dim0_stride lo32 (16,646,144)
            0,                                   // stride hi16 = 0 | dim1_stride lo16 = 0
            0                                    // dim1_stride hi = 0
        };
        v4i g2 = {0, 0, 0, 0};                   // 2D tensor: groups 2/3 unused
        v4i g3 = {0, 0, 0, 0};
        v8i g4 = {0, 0, 0, 0, 0, 0, 0, 0};       // trailing group (unused, zero)

        __builtin_amdgcn_tensor_load_to_lds(g0, g1, g2, g3, g4, 0);
    }

    // ---- Overlap with the DMA: iteration-0 jones loads (L2-resident). ----
    const int a1 = ant1[v];                      // block-uniform
    const int a2 = ant2[v];
    const float* __restrict__ j1b = jones + (size_t)a1 * ROW_F;
    const float* __restrict__ j2b = jones + (size_t)a2 * ROW_F;
    const int fo0 = 4 * threadIdx.x;

    fvec4 J1a[4], J2a[4];
    #pragma unroll
    for (int q = 0; q < 4; ++q) {
        J1a[q] = *(const fvec4*)(j1b + (size_t)q * J_PLANE + fo0);
        J2a[q] = *(const fvec4*)(j2b + (size_t)q * J_PLANE + fo0);
    }

    if (threadIdx.x < 32)
        __builtin_amdgcn_s_wait_tensorcnt(0);    // TENSORcnt == 0 -> LDS populated
    __syncthreads();

    // The compiler cannot see that the TDM writes smem: escape the LDS address
    // into a side-effecting asm so the ds_loads below are not folded away,
    // but keep the actual reads on the __shared__ array (addrspace(3)) so
    // they lower to ds_load_b128, not flat_load.
    asm volatile("" : : "s"(&smem[0]) : "memory");

    float* const ob0 = out + (size_t)v * ROW_F;

    #pragma unroll
    for (int it = 0; it < 2; ++it) {
        const int p  = threadIdx.x + it * 256;   // freq-pair index 0..511
        const int fo = 4 * p;                    // float offset in a 2048-float row

        // V planes from LDS (ds_load_b128: freqs 2p, 2p+1 as re/im pairs)
        fvec4 Vc[4];
        #pragma unroll
        for (int q = 0; q < 4; ++q)
            Vc[q] = *(const fvec4*)&smem[q * ROW_F + fo];

        // jones planes: prefetched for it=0, loaded in-loop for it=1
        fvec4 J1[4], J2[4];
        #pragma unroll
        for (int q = 0; q < 4; ++q) {
            if (it == 0) { J1[q] = J1a[q]; J2[q] = J2a[q]; }
            else {
                J1[q] = *(const fvec4*)(j1b + (size_t)q * J_PLANE + fo);
                J2[q] = *(const fvec4*)(j2b + (size_t)q * J_PLANE + fo);
            }
        }

        fvec4 O[4];
        #pragma unroll
        for (int h = 0; h < 2; ++h) {            // two freqs per float4
            fvec2 j1v[4], j2v[4], vv[4];
            #pragma unroll
            for (int q = 0; q < 4; ++q) {
                j1v[q] = h ? J1[q].hi : J1[q].lo;
                j2v[q] = h ? J2[q].hi : J2[q].lo;
                vv[q]  = h ? Vc[q].hi : Vc[q].lo;
            }
            // T[a][d] = sum_c j1[a][c] * V[c][d]
            fvec2 T[4];
            #pragma unroll
            for (int a = 0; a < 2; ++a)
                #pragma unroll
                for (int d = 0; d < 2; ++d)
                    T[a * 2 + d] = cadd(cmul(j1v[a * 2 + 0], vv[0 * 2 + d]),
                                        cmul(j1v[a * 2 + 1], vv[1 * 2 + d]));
            // O[a][b] = sum_d T[a][d] * conj(j2[b][d])
            #pragma unroll
            for (int a = 0; a < 2; ++a)
                #pragma unroll
                for (int b = 0; b < 2; ++b) {
                    fvec2 o = cadd(cmulc(T[a * 2 + 0], j2v[b * 2 + 0]),
                                   cmulc(T[a * 2 + 1], j2v[b * 2 + 1]));
                    if (h == 0) O[a * 2 + b].lo = o;
                    else        O[a * 2 + b].hi = o;
                }
        }

        // streamed output: non-temporal b128 stores (don't evict jones from L2)
        #pragma unroll
        for (int ab = 0; ab < 4; ++ab)
            __builtin_nontemporal_store(O[ab],
                (fvec4*)(ob0 + (size_t)ab * V_PLANE + fo));
    }
}

extern "C" void kernel_launch(void* const* d_in, const int* in_sizes, int n_in,
                              void* d_out, int out_size, void* d_ws, size_t ws_size,
                              hipStream_t stream) {
    const float* V     = (const float*)d_in[0];
    const float* jones = (const float*)d_in[1];
    const int*   ant1  = (const int*)d_in[2];
    const int*   ant2  = (const int*)d_in[3];
    float*       out   = (float*)d_out;
    (void)in_sizes; (void)n_in; (void)out_size; (void)d_ws; (void)ws_size;

    jones_apply_kernel<<<NVIS, 256, 0, stream>>>(V, jones, ant1, ant2, out);
}